// SSKernelNPLR_19877108646488
// MI455X (gfx1250) — compile-verified
//
#include <hip/hip_runtime.h>
#include <math.h>

// ---------------------------------------------------------------------------
// S4 NPLR kernel generation for MI455X (gfx1250, wave32, WMMA).
//   H=256 channels, N=64 half-states, RANK=1, L=8192, LF=4097.
// Pipeline: consts (z/z^2/g + twiddle table) -> prep (per-h s,m,A-rows,dt)
//        -> main: f32 WMMA Cauchy contraction + Woodbury + packed irfft.
// Fast-path notes: 1-ulp v_rcp_f32 instead of IEEE divide in the hot loop;
// FFT twiddles come from a double-built e^{i*pi*j/4096} table (no sincos).
// ---------------------------------------------------------------------------

typedef __attribute__((ext_vector_type(2))) float v2f;
typedef __attribute__((ext_vector_type(8))) float v8f;

#define H_     256
#define N_     64
#define L_     8192
#define LF_    4097
#define MHALF  4096

static __device__ constexpr double PI_D = 3.14159265358979323846;

// ---- workspace layout (float offsets) -------------------------------------
#define OFF_ZR   0
#define OFF_ZI   (LF_)
#define OFF_Z2R  (2 * LF_)
#define OFF_Z2I  (3 * LF_)
#define OFF_GR   (4 * LF_)
#define OFF_GI   (5 * LF_)          // ends at 24582
#define OFF_DT   24592              // H floats
#define OFF_S    24848              // H*N
#define OFF_M    41232              // H*N
#define OFF_A    57616              // H*8*N (rows 0..3 = Re(v_ab), 4..7 = Re(v_ab*conj(wd)))
#define OFF_TW   188688             // 4096 complex interleaved: e^{+i*pi*j/4096}
// total 196880 floats (~788 KB)

// ---- LDS layout in main kernel (float offsets) ----------------------------
#define LDS_KF   0                  // 4097 complex interleaved (8194)
#define LDS_BA   8200               // FFT buffer A: 4096 complex (8192)
#define LDS_BB   16392              // FFT buffer B
#define LDS_S    24584              // 64
#define LDS_M    24648              // 64
#define LDS_A    24712              // 8*64
#define LDS_DT   25224
#define LDS_TW   25232              // 4096 complex interleaved (8192)
#define LDS_FLOATS 33424            // 133696 bytes dynamic LDS

// ===========================================================================
// Kernel 1: per-frequency constants in double (keeps the l=L/2 node's tiny
// imaginary part so z stays finite, like the reference's float64 build).
// z = 2(1-nodes)/(1+nodes), g = 2/(1+nodes), z^2 for den = z^2 - s*z + m,
// plus the FFT twiddle table TW[j] = e^{+i*pi*j/4096}.
// ===========================================================================
__global__ void s4_consts_kernel(float* __restrict__ ws) {
    int l = blockIdx.x * blockDim.x + threadIdx.x;
    if (l >= LF_) return;
    double ang = -2.0 * PI_D * (double)l / (double)L_;
    double nr = cos(ang), ni = sin(ang);
    double dr = 1.0 + nr, di = ni;                 // 1 + nodes
    double mag = dr * dr + di * di;
    double numr = 2.0 * (1.0 - nr), numi = -2.0 * ni;
    double zr = (numr * dr + numi * di) / mag;
    double zi = (numi * dr - numr * di) / mag;
    ws[OFF_ZR  + l] = (float)zr;
    ws[OFF_ZI  + l] = (float)zi;
    ws[OFF_Z2R + l] = (float)(zr * zr - zi * zi);
    ws[OFF_Z2I + l] = (float)(2.0 * zr * zi);
    ws[OFF_GR  + l] = (float)( 2.0 * dr / mag);    // 2/(1+nodes)
    ws[OFF_GI  + l] = (float)(-2.0 * di / mag);
    if (l < MHALF) {
        double tang = PI_D * (double)l / (double)MHALF;
        ws[OFF_TW + 2 * l]     = (float)cos(tang);
        ws[OFF_TW + 2 * l + 1] = (float)sin(tang);
    }
}

// ===========================================================================
// Kernel 2: per-(h,n) prep.  wd = w*dt; s = 2Re(wd); m = |wd|^2.
// A rows (a*2+b): Re(B_a * conj(C_b)) and Re(v * conj(wd)).
// ===========================================================================
__global__ void s4_prep_kernel(const float* __restrict__ log_dt,
                               const float* __restrict__ w_re, const float* __restrict__ w_im,
                               const float* __restrict__ p_re, const float* __restrict__ p_im,
                               const float* __restrict__ q_re, const float* __restrict__ q_im,
                               const float* __restrict__ B_re, const float* __restrict__ B_im,
                               const float* __restrict__ C_re, const float* __restrict__ C_im,
                               float* __restrict__ ws) {
    int h = blockIdx.x, n = threadIdx.x;           // 64 threads
    float dtv = expf(log_dt[h]);
    float wr = w_re[h * N_ + n] * dtv;
    float wi = w_im[h * N_ + n] * dtv;
    ws[OFF_S + h * N_ + n] = 2.0f * wr;
    ws[OFF_M + h * N_ + n] = wr * wr + wi * wi;
    if (n == 0) ws[OFF_DT + h] = dtv;

    float ar[2] = { B_re[h * N_ + n], p_re[n] };
    float ai[2] = { B_im[h * N_ + n], p_im[n] };
    float cr[2] = { C_re[h * N_ + n], q_re[n] };
    float ci[2] = { C_im[h * N_ + n], q_im[n] };
#pragma unroll
    for (int a = 0; a < 2; ++a)
#pragma unroll
        for (int b = 0; b < 2; ++b) {
            float vre = ar[a] * cr[b] + ai[a] * ci[b];     // Re(Ba * conj(Cb))
            float vim = ai[a] * cr[b] - ar[a] * ci[b];
            int r = a * 2 + b;
            ws[OFF_A + (h * 8 + r)     * N_ + n] = vre;
            ws[OFF_A + (h * 8 + 4 + r) * N_ + n] = vre * wr + vim * wi;  // Re(v*conj(wd))
        }
}

// ===========================================================================
// Kernel 3: one block per h.  8 waves tile the 4097 frequencies (16/tile).
// Per tile: K=64 contraction via 16 x (2 WMMA f32 16x16x4) — A rows 0..7 are
// the per-h real coefficient rows, B fragments are inv (re/im) built in VALU.
// Lanes 0..15 then apply z-factoring + Woodbury + g and write k_f to LDS.
// Then the whole block does the packed inverse real FFT (Stockham radix-2,
// length 4096 complex, table twiddles) and writes the 8192 real samples.
// ===========================================================================
__global__ void __launch_bounds__(256)
s4_cauchy_irfft_kernel(const float* __restrict__ ws, float* __restrict__ out) {
    extern __shared__ float lds[];
    const int h    = blockIdx.x;
    const int tid  = threadIdx.x;
    const int lane = tid & 31;
    const int wid  = tid >> 5;

    for (int i = tid; i < N_; i += 256) {
        lds[LDS_S + i] = ws[OFF_S + h * N_ + i];
        lds[LDS_M + i] = ws[OFF_M + h * N_ + i];
    }
    for (int i = tid; i < 8 * N_; i += 256) lds[LDS_A + i] = ws[OFF_A + h * 8 * N_ + i];
    for (int i = tid; i < 2 * MHALF; i += 256) lds[LDS_TW + i] = ws[OFF_TW + i];
    if (tid == 0) lds[LDS_DT] = ws[OFF_DT + h];
    __syncthreads();

    // A fragments: 16x4 f32 -> lane m = lane&15 holds row m; lanes>=16 take
    // K pair {2,3}.  Rows 8..15 padded with zero.
    const int mrow = lane & 15;
    const int kb   = 2 * (lane >> 4);
    v2f afrag[16];
#pragma unroll
    for (int kk = 0; kk < 16; ++kk) {
        int nb = 4 * kk + kb;
        float ax = 0.0f, ay = 0.0f;
        if (mrow < 8) { ax = lds[LDS_A + mrow * N_ + nb]; ay = lds[LDS_A + mrow * N_ + nb + 1]; }
        afrag[kk].x = ax; afrag[kk].y = ay;
    }
    const float dtv = lds[LDS_DT];

    for (int tile = wid; tile < 257; tile += 8) {          // 257*16 covers 4097
        int l  = tile * 16 + mrow;                         // this lane's frequency column
        int lc = (l > MHALF) ? MHALF : l;                  // clamp pad columns
        float zr  = ws[OFF_ZR  + lc], zi  = ws[OFF_ZI  + lc];
        float z2r = ws[OFF_Z2R + lc], z2i = ws[OFF_Z2I + lc];

        v8f acr = {0.f,0.f,0.f,0.f,0.f,0.f,0.f,0.f};       // Sum A*inv_re
        v8f aci = {0.f,0.f,0.f,0.f,0.f,0.f,0.f,0.f};       // Sum A*inv_im
#pragma unroll
        for (int kk = 0; kk < 16; ++kk) {
            int nb = 4 * kk + kb;
            // inv = 1/den, den = z^2 - s*z + m  (== (z-wd)(z-conj wd))
            float s0 = lds[LDS_S + nb],     m0 = lds[LDS_M + nb];
            float s1 = lds[LDS_S + nb + 1], m1 = lds[LDS_M + nb + 1];
            float dr0 = z2r - s0 * zr + m0, di0 = z2i - s0 * zi;
            float dr1 = z2r - s1 * zr + m1, di1 = z2i - s1 * zi;
            float rm0 = __builtin_amdgcn_rcpf(dr0 * dr0 + di0 * di0);  // v_rcp_f32
            float rm1 = __builtin_amdgcn_rcpf(dr1 * dr1 + di1 * di1);
            v2f bre, bim;
            bre.x =  dr0 * rm0;  bim.x = -di0 * rm0;
            bre.y =  dr1 * rm1;  bim.y = -di1 * rm1;
            acr = __builtin_amdgcn_wmma_f32_16x16x4_f32(false, afrag[kk], false, bre,
                                                        (short)0, acr, false, false);
            aci = __builtin_amdgcn_wmma_f32_16x16x4_f32(false, afrag[kk], false, bim,
                                                        (short)0, aci, false, false);
        }

        if (lane < 16 && l <= MHALF) {                     // D rows 0..7 live in lanes 0..15
            float sc = 2.0f * dtv;
            float rr[4], ri[4];
#pragma unroll
            for (int ab = 0; ab < 4; ++ab) {
                float s1r = acr[ab],     s1i = aci[ab];     // Sum v_r * inv
                float s2r = acr[4 + ab], s2i = aci[4 + ab]; // Sum vw_r * inv
                rr[ab] = sc * (zr * s1r - zi * s1i - s2r);  // 2*dt*(z*S1 - S2)
                ri[ab] = sc * (zr * s1i + zi * s1r - s2i);
            }
            // Woodbury: k_f = r00 - r01*r10/(1+r11)
            float wre = rr[1] * rr[2] - ri[1] * ri[2];
            float wim = rr[1] * ri[2] + ri[1] * rr[2];
            float dre = 1.0f + rr[3], dim = ri[3];
            float rmg = __builtin_amdgcn_rcpf(dre * dre + dim * dim);
            float cre = (wre * dre + wim * dim) * rmg;
            float cim = (wim * dre - wre * dim) * rmg;
            float kre = rr[0] - cre, kim = ri[0] - cim;
            float gr = ws[OFF_GR + l], gi = ws[OFF_GI + l]; // * 2/(1+nodes)
            lds[LDS_KF + 2 * l]     = kre * gr - kim * gi;
            lds[LDS_KF + 2 * l + 1] = kre * gi + kim * gr;
        }
    }
    __syncthreads();

    // ---- packed irfft: K[0..4096] -> Z[0..4095], complex IFFT(4096) -------
    // E = (K[m]+conj(K[M-m]))/2 ; O = (K[m]-conj(K[M-m]))/2 * e^{+i*pi*m/M}
    // Z = E + i*O ; x[2j]=Re(ifft(Z))[j], x[2j+1]=Im(...)
    for (int m = tid; m < MHALF; m += 256) {
        float K1r = lds[LDS_KF + 2 * m],  K1i = lds[LDS_KF + 2 * m + 1];
        int   m2  = MHALF - m;
        float K2r = lds[LDS_KF + 2 * m2], K2i = lds[LDS_KF + 2 * m2 + 1];
        float Er = 0.5f * (K1r + K2r), Ei = 0.5f * (K1i - K2i);
        float Dr = 0.5f * (K1r - K2r), Di = 0.5f * (K1i + K2i);
        float cs = lds[LDS_TW + 2 * m], sn = lds[LDS_TW + 2 * m + 1];
        float Or = Dr * cs - Di * sn;
        float Oi = Dr * sn + Di * cs;
        lds[LDS_BA + 2 * m]     = Er - Oi;                  // Z = E + i*O
        lds[LDS_BA + 2 * m + 1] = Ei + Or;
    }
    __syncthreads();

    // ---- 12 Stockham radix-2 inverse stages (length 4096) -----------------
    // twiddle e^{+i*pi*p/nh} = TW[p * 4096/nh] = TW[p << (sh+1)]
    int xb = LDS_BA, yb = LDS_BB;
    int sh = 0;
    for (int nh = 2048; nh >= 1; nh >>= 1, ++sh) {
        int ssz = 1 << sh;
        for (int t = tid; t < 2048; t += 256) {
            int p = t >> sh;
            int q = t & (ssz - 1);
            int ia = q + ssz * p;
            int ib = q + ssz * (p + nh);
            float ar = lds[xb + 2 * ia], ai2 = lds[xb + 2 * ia + 1];
            float br = lds[xb + 2 * ib], bi2 = lds[xb + 2 * ib + 1];
            int j = p << (sh + 1);
            float cs = lds[LDS_TW + 2 * j], sn = lds[LDS_TW + 2 * j + 1];
            int io = q + ssz * 2 * p;
            lds[yb + 2 * io]     = ar + br;
            lds[yb + 2 * io + 1] = ai2 + bi2;
            float dr2 = ar - br, di2 = ai2 - bi2;
            int io2 = io + ssz;
            lds[yb + 2 * io2]     = dr2 * cs - di2 * sn;
            lds[yb + 2 * io2 + 1] = dr2 * sn + di2 * cs;
        }
        int tmp = xb; xb = yb; yb = tmp;
        __syncthreads();
    }

    const float scl = 1.0f / (float)MHALF;                  // ifft 1/M
    for (int j = tid; j < MHALF; j += 256) {
        out[h * L_ + 2 * j]     = lds[xb + 2 * j]     * scl;
        out[h * L_ + 2 * j + 1] = lds[xb + 2 * j + 1] * scl;
    }
}

// ===========================================================================
extern "C" void kernel_launch(void* const* d_in, const int* in_sizes, int n_in,
                              void* d_out, int out_size, void* d_ws, size_t ws_size,
                              hipStream_t stream) {
    const float* log_dt = (const float*)d_in[0];
    const float* w_re   = (const float*)d_in[1];
    const float* w_im   = (const float*)d_in[2];
    const float* p_re   = (const float*)d_in[3];
    const float* p_im   = (const float*)d_in[4];
    const float* q_re   = (const float*)d_in[5];
    const float* q_im   = (const float*)d_in[6];
    const float* B_re   = (const float*)d_in[7];
    const float* B_im   = (const float*)d_in[8];
    const float* C_re   = (const float*)d_in[9];
    const float* C_im   = (const float*)d_in[10];
    float* ws  = (float*)d_ws;
    float* outp = (float*)d_out;

    s4_consts_kernel<<<(LF_ + 255) / 256, 256, 0, stream>>>(ws);
    s4_prep_kernel<<<H_, N_, 0, stream>>>(log_dt, w_re, w_im, p_re, p_im,
                                          q_re, q_im, B_re, B_im, C_re, C_im, ws);
    s4_cauchy_irfft_kernel<<<H_, 256, LDS_FLOATS * 4, stream>>>(ws, outp);
}